// DRKD_41266045780703
// MI455X (gfx1250) — compile-verified
//
#include <hip/hip_runtime.h>
#include <math.h>

typedef __attribute__((ext_vector_type(2))) float v2f;
typedef __attribute__((ext_vector_type(8))) float v8f;

#define N 512
#define Dn 128
#define NC 8
#define N2 (N * N)

// ---- workspace layout (floats) ----
// [0)          G_s   : N2
// [N2)         G_t   : N2
// [2*N2)       invn_s: N2
// [3*N2)       invn_t: N2
// [4*N2)       partial_dist : 1024
// [+1024)      partial_angle: 2048
// [+2048)      stat partials: sum_s[1024] cnt_s[1024] sum_t[1024] cnt_t[1024]
// [+4096)      scalars[8]: 0=inv_mean_s 1=inv_mean_t 2=ce_wnll 3=ce_wsum
#define OFF_GS   0
#define OFF_GT   (N2)
#define OFF_WS_S (2 * N2)
#define OFF_WS_T (3 * N2)
#define OFF_PD   (4 * N2)
#define OFF_PA   (OFF_PD + 1024)
#define OFF_ST   (OFF_PA + 2048)
#define OFF_SC   (OFF_ST + 4096)

__device__ __forceinline__ float block_reduce_256(float v, float* sm) {
    const int tid = threadIdx.x;
    sm[tid] = v;
    __syncthreads();
    #pragma unroll
    for (int s = 128; s > 0; s >>= 1) {
        if (tid < s) sm[tid] += sm[tid + s];
        __syncthreads();
    }
    float r = sm[0];
    __syncthreads();
    return r;
}

__device__ __forceinline__ float smooth_l1(float a, float b) {
    float d = fabsf(a - b);
    return (d < 1.0f) ? 0.5f * d * d : d - 0.5f;
}

// ---------------- K1: Gram matrices via f32 WMMA -----------------------------
// 2048 tiles total (1024 per matrix); 8 waves per block -> 256 blocks.
__global__ void gram_kernel(const float* __restrict__ Fs,
                            const float* __restrict__ Ft,
                            float* __restrict__ Gs,
                            float* __restrict__ Gt) {
    const int lane = threadIdx.x & 31;
    const int wave = threadIdx.x >> 5;
    const int gw   = blockIdx.x * 8 + wave;       // 0..2047 (wave-uniform)
    const float* F = (gw < 1024) ? Fs : Ft;
    float*       G = (gw < 1024) ? Gs : Gt;
    const int tile = gw & 1023;
    const int m0 = (tile >> 5) << 4;
    const int n0 = (tile & 31) << 4;

    // ISA f32 A/B layout: lanes 0-15 -> rows M/N 0..15 with K=(0,1);
    // lanes 16-31 -> same rows with K=(2,3). 2 fp32 per lane per operand.
    const int r  = lane & 15;
    const int kb = (lane >> 4) << 1;
    const float* arow = F + (m0 + r) * Dn + kb;
    const float* brow = F + (n0 + r) * Dn + kb;

    v8f acc = {};
    #pragma unroll 4
    for (int kk = 0; kk < Dn; kk += 4) {
        v2f a = *(const v2f*)(arow + kk);
        v2f b = *(const v2f*)(brow + kk);
        acc = __builtin_amdgcn_wmma_f32_16x16x4_f32(
            false, a, false, b, (short)0, acc, false, false);
    }

    // C/D layout: VGPR rr -> M = rr (lanes 0-15) or rr+8 (lanes 16-31), N = lane&15
    const int hi  = lane >> 4;
    const int col = n0 + (lane & 15);
    #pragma unroll
    for (int rr = 0; rr < 8; ++rr) {
        G[(m0 + rr + 8 * hi) * N + col] = acc[rr];
    }
}

// ---------------- K2: per-pair stats (invn matrices + mean_pos partials) -----
__global__ void stats_kernel(const float* __restrict__ Gs,
                             const float* __restrict__ Gt,
                             float* __restrict__ W) {
    __shared__ float sm[256];
    const int tid = threadIdx.x;
    const int idx = blockIdx.x * 256 + tid;     // 0..N2-1
    const int i = idx >> 9, j = idx & (N - 1);

    float gs  = Gs[idx];
    float vs  = Gs[i * (N + 1)] + Gs[j * (N + 1)] - 2.0f * gs;
    float ns  = sqrtf(fmaxf(vs, 0.0f));
    W[OFF_WS_S + idx] = 1.0f / fmaxf(ns, 1e-12f);
    float dsv = (i == j) ? 0.0f : sqrtf(fmaxf(vs, 1e-12f));

    float gt  = Gt[idx];
    float vt  = Gt[i * (N + 1)] + Gt[j * (N + 1)] - 2.0f * gt;
    float nt  = sqrtf(fmaxf(vt, 0.0f));
    W[OFF_WS_T + idx] = 1.0f / fmaxf(nt, 1e-12f);
    float dtv = (i == j) ? 0.0f : sqrtf(fmaxf(vt, 1e-12f));

    float r;
    r = block_reduce_256(dsv, sm);
    if (tid == 0) W[OFF_ST + 0 * 1024 + blockIdx.x] = r;
    r = block_reduce_256((dsv > 0.0f) ? 1.0f : 0.0f, sm);
    if (tid == 0) W[OFF_ST + 1 * 1024 + blockIdx.x] = r;
    r = block_reduce_256(dtv, sm);
    if (tid == 0) W[OFF_ST + 2 * 1024 + blockIdx.x] = r;
    r = block_reduce_256((dtv > 0.0f) ? 1.0f : 0.0f, sm);
    if (tid == 0) W[OFF_ST + 3 * 1024 + blockIdx.x] = r;
}

// ---------------- K2b: fold stat partials -> inverse means -------------------
__global__ void means_kernel(float* __restrict__ W) {
    __shared__ float sm[256];
    const int tid = threadIdx.x;
    float acc[4] = {0.f, 0.f, 0.f, 0.f};
    for (int t = tid; t < 1024; t += 256) {
        acc[0] += W[OFF_ST + 0 * 1024 + t];
        acc[1] += W[OFF_ST + 1 * 1024 + t];
        acc[2] += W[OFF_ST + 2 * 1024 + t];
        acc[3] += W[OFF_ST + 3 * 1024 + t];
    }
    float sum_s = block_reduce_256(acc[0], sm);
    float cnt_s = block_reduce_256(acc[1], sm);
    float sum_t = block_reduce_256(acc[2], sm);
    float cnt_t = block_reduce_256(acc[3], sm);
    if (tid == 0) {
        W[OFF_SC + 0] = cnt_s / sum_s;   // 1 / mean_pos(d_s)
        W[OFF_SC + 1] = cnt_t / sum_t;   // 1 / mean_pos(d_t)
    }
}

// ---------------- K3: RKD distance loss partials -----------------------------
__global__ void dist_kernel(const float* __restrict__ Gs,
                            const float* __restrict__ Gt,
                            float* __restrict__ W) {
    __shared__ float sm[256];
    const int tid = threadIdx.x;
    const int idx = blockIdx.x * 256 + tid;
    const int i = idx >> 9, j = idx & (N - 1);
    const float im_s = W[OFF_SC + 0];
    const float im_t = W[OFF_SC + 1];

    float vs  = Gs[i * (N + 1)] + Gs[j * (N + 1)] - 2.0f * Gs[idx];
    float dsv = (i == j) ? 0.0f : sqrtf(fmaxf(vs, 1e-12f));
    float vt  = Gt[i * (N + 1)] + Gt[j * (N + 1)] - 2.0f * Gt[idx];
    float dtv = (i == j) ? 0.0f : sqrtf(fmaxf(vt, 1e-12f));

    float l = smooth_l1(dsv * im_s, dtv * im_t);
    float r = block_reduce_256(l, sm);
    if (tid == 0) W[OFF_PD + blockIdx.x] = r;
}

// ---------------- K4: RKD angle loss -----------------------------------------
// 2048 blocks: block handles i = blockIdx>>2 and a 128-row j-segment.
// Threads: (tid>>7) selects one of 2 rows per iteration, (tid&127)*4 is the
// float4 k-chunk -> global_load_b128 + ds_load_b128 in the hot loop.
__global__ void angle_kernel(const float* __restrict__ Gs,
                             const float* __restrict__ Gt,
                             const float* __restrict__ Ws,
                             const float* __restrict__ Wt,
                             float* __restrict__ W) {
    __shared__ float rGs[N], rGt[N], rWs[N], rWt[N];
    __shared__ float t1s[N], t1t[N];
    __shared__ float sm[256];
    const int tid  = threadIdx.x;
    const int i    = blockIdx.x >> 2;
    const int jseg = (blockIdx.x & 3) << 7;      // 0,128,256,384

    for (int t = tid; t < N; t += 256) {
        rGs[t] = Gs[i * N + t];
        rGt[t] = Gt[i * N + t];
        rWs[t] = Ws[i * N + t];
        rWt[t] = Wt[i * N + t];
    }
    __syncthreads();
    const float gsii = rGs[i];
    const float gtii = rGt[i];
    for (int t = tid; t < N; t += 256) {
        t1s[t] = rGs[t] - gsii;     // G[i,j] - G[i,i]
        t1t[t] = rGt[t] - gtii;
    }
    __syncthreads();

    const int jof = tid >> 7;            // 0 or 1
    const int kv  = (tid & 127) << 2;    // k chunk base: 0..508

    float acc = 0.0f;
    for (int j0 = jseg; j0 < jseg + 128; j0 += 2) {
        const int j = j0 + jof;
        const float t1sj = t1s[j];
        const float t1tj = t1t[j];
        const float wsj  = rWs[j];
        const float wtj  = rWt[j];

        const float4 g4s = *(const float4*)(Gs + j * N + kv);
        const float4 g4t = *(const float4*)(Gt + j * N + kv);
        const float4 gks = *(const float4*)(&rGs[kv]);
        const float4 gkt = *(const float4*)(&rGt[kv]);
        const float4 wks = *(const float4*)(&rWs[kv]);
        const float4 wkt = *(const float4*)(&rWt[kv]);

        acc += smooth_l1(((g4s.x - gks.x) - t1sj) * (wsj * wks.x),
                         ((g4t.x - gkt.x) - t1tj) * (wtj * wkt.x));
        acc += smooth_l1(((g4s.y - gks.y) - t1sj) * (wsj * wks.y),
                         ((g4t.y - gkt.y) - t1tj) * (wtj * wkt.y));
        acc += smooth_l1(((g4s.z - gks.z) - t1sj) * (wsj * wks.z),
                         ((g4t.z - gkt.z) - t1tj) * (wtj * wkt.z));
        acc += smooth_l1(((g4s.w - gks.w) - t1sj) * (wsj * wks.w),
                         ((g4t.w - gkt.w) - t1tj) * (wtj * wkt.w));
    }
    float r = block_reduce_256(acc, sm);
    if (tid == 0) W[OFF_PA + blockIdx.x] = r;
}

// ---------------- K5: weighted cross-entropy ---------------------------------
__global__ void ce_kernel(const float* __restrict__ logits,
                          const float* __restrict__ weight,
                          const int* __restrict__ target,
                          float* __restrict__ W) {
    __shared__ float sm[256];
    const int tid = threadIdx.x;
    float wnll = 0.0f, wsum = 0.0f;
    for (int s = tid; s < N; s += 256) {
        const float* l = logits + s * NC;
        int tg = target[s];
        float m = l[0];
        #pragma unroll
        for (int c = 1; c < NC; ++c) m = fmaxf(m, l[c]);
        float se = 0.0f;
        #pragma unroll
        for (int c = 0; c < NC; ++c) se += __expf(l[c] - m);
        float lse = logf(se) + m;
        float w = weight[tg];
        wnll += w * (lse - l[tg]);
        wsum += w;
    }
    float a = block_reduce_256(wnll, sm);
    float b = block_reduce_256(wsum, sm);
    if (tid == 0) {
        W[OFF_SC + 2] = a;
        W[OFF_SC + 3] = b;
    }
}

// ---------------- K6: finalize ----------------------------------------------
__global__ void finalize_kernel(const float* __restrict__ W, float* __restrict__ out) {
    __shared__ float sm[256];
    const int tid = threadIdx.x;
    float sd = 0.0f, sa = 0.0f;
    for (int t = tid; t < 1024; t += 256) sd += W[OFF_PD + t];
    for (int t = tid; t < 2048; t += 256) sa += W[OFF_PA + t];
    float dist_sum  = block_reduce_256(sd, sm);
    float angle_sum = block_reduce_256(sa, sm);
    if (tid == 0) {
        float loss_dist  = dist_sum / (float)N2;
        float loss_angle = angle_sum / ((float)N * (float)N * (float)N);
        float loss_ce    = W[OFF_SC + 2] / W[OFF_SC + 3];
        out[0] = 25.0f * loss_dist + 50.0f * loss_angle + 3.0f * loss_ce;
    }
}

extern "C" void kernel_launch(void* const* d_in, const int* in_sizes, int n_in,
                              void* d_out, int out_size, void* d_ws, size_t ws_size,
                              hipStream_t stream) {
    const float* feat_s   = (const float*)d_in[0];
    const float* feat_t   = (const float*)d_in[1];
    const float* logits_s = (const float*)d_in[2];
    const float* weight   = (const float*)d_in[3];
    const int*   target   = (const int*)d_in[4];
    float* out = (float*)d_out;
    float* W   = (float*)d_ws;

    float* Gs  = W + OFF_GS;
    float* Gt  = W + OFF_GT;
    float* Ws_ = W + OFF_WS_S;
    float* Wt_ = W + OFF_WS_T;

    gram_kernel<<<256, 256, 0, stream>>>(feat_s, feat_t, Gs, Gt);
    stats_kernel<<<1024, 256, 0, stream>>>(Gs, Gt, W);
    means_kernel<<<1, 256, 0, stream>>>(W);
    dist_kernel<<<1024, 256, 0, stream>>>(Gs, Gt, W);
    angle_kernel<<<2048, 256, 0, stream>>>(Gs, Gt, Ws_, Wt_, W);
    ce_kernel<<<1, 256, 0, stream>>>(logits_s, weight, target, W);
    finalize_kernel<<<1, 256, 0, stream>>>(W, out);
}